// ClassModel_72318659330834
// MI455X (gfx1250) — compile-verified
//
#include <hip/hip_runtime.h>
#include <hip/hip_bf16.h>
#include <math.h>

#define A_N    128
#define S_N    512
#define D_N    768
#define NL_N   10
#define LW_N   5
#define MASKID 103
#define NEGBIG (-1.0e30f)

#define CH     8                 // rows of bert staged per chunk
#define NCH    (S_N / CH)        // 64 chunks

typedef __attribute__((ext_vector_type(2))) float v2f;
typedef __attribute__((ext_vector_type(8))) float v8f;
typedef __attribute__((ext_vector_type(4))) int   v4i;

#if defined(__HIP_DEVICE_COMPILE__) && \
    __has_builtin(__builtin_amdgcn_global_load_async_to_lds_b128) && \
    __has_builtin(__builtin_amdgcn_s_wait_asynccnt)
#define HAVE_ASYNC 1
#else
#define HAVE_ASYNC 0
#endif

#if HAVE_ASYNC
typedef __attribute__((address_space(1))) v4i g_v4i;   // global 16B vector
typedef __attribute__((address_space(3))) v4i l_v4i;   // LDS 16B vector
// 16-byte global -> LDS async copy (ASYNCcnt-tracked)
#define ASYNC_CP16(gsrc, ldst)                                              \
    __builtin_amdgcn_global_load_async_to_lds_b128(                         \
        (g_v4i*)(gsrc), (l_v4i*)(ldst), 0, 0)
#endif

// ---------------------------------------------------------------------------
// Kernel 1: per-sample setup: mask_pos scan, mask_logit copy into x[:,768:],
// senti_out = bert[:,0] @ W_senti.T + b_senti
// ---------------------------------------------------------------------------
__global__ void k_setup(const float* __restrict__ bert,
                        const int*   __restrict__ ids,
                        const float* __restrict__ Ws,
                        const float* __restrict__ bs,
                        float*       __restrict__ x,     // [A, 1536]
                        float*       __restrict__ out)   // senti at [0..255]
{
    const int a = blockIdx.x;
    const int t = threadIdx.x;   // 256 threads
    __shared__ int   mp_sh;
    __shared__ float red0[256];
    __shared__ float red1[256];

    if (t == 0) {
        int mp = 0;
        const int* row = ids + a * S_N;
        for (int s = 0; s < S_N; ++s) {
            if (row[s] == MASKID) { mp = s; break; }
        }
        mp_sh = mp;
    }
    __syncthreads();
    const int mp = mp_sh;

    const float* mrow = bert + ((size_t)a * S_N + mp) * D_N;
    const float* c0   = bert + (size_t)a * S_N * D_N;     // s = 0 row
    float* xr = x + (size_t)a * (2 * D_N) + D_N;

    float p0 = 0.f, p1 = 0.f;
    for (int i = t; i < D_N; i += 256) {
        xr[i] = mrow[i];                 // mask_logit -> x[:, 768:]
        float v = c0[i];
        p0 += v * Ws[i];                 // W_senti[0][i]
        p1 += v * Ws[D_N + i];           // W_senti[1][i]
    }
    red0[t] = p0; red1[t] = p1;
    __syncthreads();
    for (int off = 128; off > 0; off >>= 1) {
        if (t < off) { red0[t] += red0[t + off]; red1[t] += red1[t + off]; }
        __syncthreads();
    }
    if (t == 0) {
        out[a * 2 + 0] = red0[0] + bs[0];
        out[a * 2 + 1] = red1[0] + bs[1];
    }
}

// ---------------------------------------------------------------------------
// Kernel 2: single-pass masked-softmax attention with online rescaling.
// bert[a] is streamed ONCE through LDS via double-buffered async copies:
//   chunk (8 rows, 24KB) -> LDS buf[c&1] while buf[(c+1)&1] is in flight.
// scores from LDS (wave-per-row, shfl reduce), att accumulated with running
// max/sum. Result att -> x[:, :768].
// ---------------------------------------------------------------------------
__global__ void __launch_bounds__(256)
k_attn(const float* __restrict__ bert,
       const int*   __restrict__ length,
       float*       __restrict__ x)      // [A, 1536]
{
    const int a    = blockIdx.x;
    const int t    = threadIdx.x;        // 256 threads = 8 waves
    const int wv   = t >> 5;
    const int lane = t & 31;

    __shared__ float buf[2][CH * D_N];   // 2 x 24KB
    __shared__ float mlog[D_N];
    __shared__ float scq[CH];
    __shared__ float eq[CH];

    const float* base = bert + (size_t)a * S_N * D_N;

    // stage mask_logit
    {
        const float* xr = x + (size_t)a * (2 * D_N) + D_N;
        for (int i = t; i < D_N; i += 256) mlog[i] = xr[i];
    }
    const int len = length[a];

#if HAVE_ASYNC
    // prologue: chunks 0 and 1 in flight (6 async instrs per wave per chunk)
    {
        const float* g0 = base;
        const float* g1 = base + (size_t)CH * D_N;
        #pragma unroll
        for (int j = 0; j < 6; ++j) {
            const int idx = t + 256 * j;             // 16B units, 0..1535
            ASYNC_CP16(g0 + idx * 4, &buf[0][idx * 4]);
        }
        #pragma unroll
        for (int j = 0; j < 6; ++j) {
            const int idx = t + 256 * j;
            ASYNC_CP16(g1 + idx * 4, &buf[1][idx * 4]);
        }
    }
#endif

    float m_run = NEGBIG, l_run = 0.f;
    float a0 = 0.f, a1 = 0.f, a2 = 0.f;
    const float4* ml4 = (const float4*)mlog;

    for (int c = 0; c < NCH; ++c) {
        const int b = c & 1;

#if HAVE_ASYNC
        // async loads complete in order: waiting to <=6 outstanding guarantees
        // chunk c landed (chunk c+1's 6 may still be in flight).
        if (c == NCH - 1) __builtin_amdgcn_s_wait_asynccnt(0);
        else              __builtin_amdgcn_s_wait_asynccnt(6);
#else
        {   // synchronous fallback copy of chunk c
            const float4* g4 = (const float4*)(base + (size_t)c * CH * D_N);
            float4* b4 = (float4*)buf[b];
            #pragma unroll
            for (int j = 0; j < 6; ++j) b4[t + 256 * j] = g4[t + 256 * j];
        }
#endif
        __syncthreads();   // chunk c visible to all waves

        // scores: wave wv owns row wv of the chunk
        {
            const float4* row4 = (const float4*)&buf[b][wv * D_N];
            float sum = 0.f;
            #pragma unroll
            for (int i = 0; i < 6; ++i) {
                float4 v = row4[lane + 32 * i];
                float4 m = ml4[lane + 32 * i];
                sum += v.x * m.x + v.y * m.y + v.z * m.z + v.w * m.w;
            }
            for (int off = 16; off > 0; off >>= 1)
                sum += __shfl_down(sum, off, 32);
            if (lane == 0) {
                const int s = c * CH + wv;
                const bool valid = (s >= 3) && (s < 3 + len);
                scq[wv] = valid ? sum : NEGBIG;
            }
        }
        __syncthreads();   // scq ready

        // online softmax bookkeeping (identical in every thread)
        float mc = scq[0];
        #pragma unroll
        for (int r = 1; r < CH; ++r) mc = fmaxf(mc, scq[r]);
        const float m_new = fmaxf(m_run, mc);
        const float scale = expf(m_run - m_new);
        if (t < CH) eq[t] = expf(scq[t] - m_new);
        m_run = m_new;
        a0 *= scale; a1 *= scale; a2 *= scale; l_run *= scale;
        __syncthreads();   // eq ready

        // accumulate att: thread t owns d = t, t+256, t+512
        #pragma unroll
        for (int r = 0; r < CH; ++r) {
            const float e = eq[r];
            l_run += e;
            const float* row = &buf[b][r * D_N];
            a0 += e * row[t];
            a1 += e * row[t + 256];
            a2 += e * row[t + 512];
        }
        __syncthreads();   // everyone done reading buf[b] before refill

#if HAVE_ASYNC
        if (c + 2 < NCH) {
            const float* g = base + (size_t)(c + 2) * CH * D_N;
            #pragma unroll
            for (int j = 0; j < 6; ++j) {
                const int idx = t + 256 * j;
                ASYNC_CP16(g + idx * 4, &buf[b][idx * 4]);
            }
        }
#endif
    }

    const float inv = 1.0f / l_run;
    float* xa = x + (size_t)a * (2 * D_N);
    xa[t]       = a0 * inv;
    xa[t + 256] = a1 * inv;
    xa[t + 512] = a2 * inv;
}

// ---------------------------------------------------------------------------
// Kernel 3: h = tanh(x @ W_dense.T + b_dense) via V_WMMA_F32_16X16X4_F32
// x: [128, 1536], W_dense: [768, 1536], h: [128, 768]
// One wave per 16x16 output tile; K-loop over 1536 in steps of 4.
// ---------------------------------------------------------------------------
__global__ void k_dense(const float* __restrict__ x,
                        const float* __restrict__ Wd,
                        const float* __restrict__ bd,
                        float*       __restrict__ h)
{
    const int n0   = blockIdx.x * 16;
    const int m0   = blockIdx.y * 16;
    const int lane = threadIdx.x;      // 32 threads = 1 wave
    const int half = lane >> 4;        // 0 or 1
    const int l16  = lane & 15;

    const float* arow = x  + (size_t)(m0 + l16) * (2 * D_N);   // A[m, :]
    const float* brow = Wd + (size_t)(n0 + l16) * (2 * D_N);   // B[:, n] = Wd[n, :]

    v8f acc = {};
    for (int kk = 0; kk < 2 * D_N; kk += 4) {
        const int o = kk + 2 * half;
        // A layout (16x4 f32): lane = 16*half + m, regs hold K = 2*half, 2*half+1
        v2f av = *(const v2f*)(arow + o);
        // B layout (4x16 f32): lane = 16*half + n, regs hold K = 2*half, 2*half+1
        v2f bv = *(const v2f*)(brow + o);
        acc = __builtin_amdgcn_wmma_f32_16x16x4_f32(
            /*neg_a=*/false, av, /*neg_b=*/false, bv,
            /*c_mod=*/(short)0, acc, /*reuse_a=*/false, /*reuse_b=*/false);
    }

    const float bias = bd[n0 + l16];
    #pragma unroll
    for (int r = 0; r < 8; ++r) {
        const int m = m0 + r + 8 * half;          // D: VGPR r <-> M = r (+8 for hi half)
        h[(size_t)m * D_N + (n0 + l16)] = tanhf(acc[r] + bias);
    }
}

// ---------------------------------------------------------------------------
// Kernel 4: lp[l,w] = tanh(h[a] . W_prob[label_words[l,w]] + b_prob[idx]);
//           out[a,k,l] = sum_w lp[l,w] * W_lab[l,k,w]
// (exact shortcut: only the 50 gathered rows of W_prob are ever needed)
// ---------------------------------------------------------------------------
__global__ void k_final(const float* __restrict__ h,     // [A, 768]
                        const float* __restrict__ Wp,    // [V, 768]
                        const float* __restrict__ bp,    // [V]
                        const int*   __restrict__ lw,    // [10*5]
                        const float* __restrict__ Wlab,  // [10,2,5]
                        float*       __restrict__ out)   // base at d_out+256
{
    const int a    = blockIdx.x;
    const int t    = threadIdx.x;    // 256 threads = 8 waves
    const int wave = t >> 5;
    const int lane = t & 31;
    __shared__ float lp[NL_N * LW_N];

    const float4* hr = (const float4*)(h + (size_t)a * D_N);
    for (int p = wave; p < NL_N * LW_N; p += 8) {
        const int idx = lw[p];
        const float4* wr = (const float4*)(Wp + (size_t)idx * D_N);
        float sum = 0.f;
        #pragma unroll
        for (int i = 0; i < 6; ++i) {           // 6*32 float4 = 768 floats
            float4 hv = hr[lane + 32 * i];
            float4 wv = wr[lane + 32 * i];
            sum += hv.x * wv.x + hv.y * wv.y + hv.z * wv.z + hv.w * wv.w;
        }
        for (int off = 16; off > 0; off >>= 1)
            sum += __shfl_down(sum, off, 32);    // wave32 reduction
        if (lane == 0) lp[p] = tanhf(sum + bp[idx]);
    }
    __syncthreads();

    if (t < 20) {
        const int k = t / 10, l = t % 10;
        float v = 0.f;
        #pragma unroll
        for (int w = 0; w < LW_N; ++w)
            v += lp[l * LW_N + w] * Wlab[l * (2 * LW_N) + k * LW_N + w];
        out[a * 20 + k * 10 + l] = v;
    }
}

// ---------------------------------------------------------------------------
extern "C" void kernel_launch(void* const* d_in, const int* in_sizes, int n_in,
                              void* d_out, int out_size, void* d_ws, size_t ws_size,
                              hipStream_t stream) {
    const float* bert  = (const float*)d_in[0];
    const int*   ids   = (const int*)  d_in[1];
    const int*   len   = (const int*)  d_in[2];
    const int*   lw    = (const int*)  d_in[3];
    const float* Ws    = (const float*)d_in[4];
    const float* bs    = (const float*)d_in[5];
    const float* Wd    = (const float*)d_in[6];
    const float* bd    = (const float*)d_in[7];
    const float* Wp    = (const float*)d_in[8];
    const float* bp    = (const float*)d_in[9];
    const float* Wlab  = (const float*)d_in[10];
    float* out = (float*)d_out;

    float* x = (float*)d_ws;                 // [128, 1536]
    float* h = x + (size_t)A_N * 2 * D_N;    // [128, 768]

    k_setup<<<dim3(A_N), dim3(256), 0, stream>>>(bert, ids, Ws, bs, x, out);
    k_attn <<<dim3(A_N), dim3(256), 0, stream>>>(bert, len, x);
    k_dense<<<dim3(D_N / 16, A_N / 16), dim3(32), 0, stream>>>(x, Wd, bd, h);
    k_final<<<dim3(A_N), dim3(256), 0, stream>>>(h, Wp, bp, lw, Wlab, out + 256);
}